// ContrastLoss_52948356825829
// MI455X (gfx1250) — compile-verified
//
#include <hip/hip_runtime.h>
#include <hip/hip_bf16.h>
#include <math.h>

// ---------------------------------------------------------------------------
// ContrastLoss: B=32, S=64, H=256.
// Memory-bound streaming reduction over tmap (128 MiB, read once, ~5.6us floor
// at 23.3 TB/s HBM; tensor even fits in the 192 MB L2 for replays). Dot
// products via V_WMMA_F32_16X16X4_F32 (f32 in/out): one wave handles 16 (i,j)
// cells; A = 16 rows x K=4 chunk of tmap, B = normalized q broadcast into all
// 16 columns. b128 loads feed two WMMAs each; the K-reduction is manually
// software-pipelined (prefetch distance 2) for ~6 loads in flight per wave.
// ---------------------------------------------------------------------------

#define CL_EPS 1e-8f

typedef float v2f __attribute__((ext_vector_type(2)));
typedef float v4f __attribute__((ext_vector_type(4)));
typedef float v8f __attribute__((ext_vector_type(8)));

constexpr int kB = 32, kS = 64, kH = 256;
constexpr int kCells          = kB * kS * kS;     // 131072
constexpr int kCellsPerBatch  = kS * kS;          // 4096
constexpr int kTiles          = kCells / 16;      // 8192 (one 16-cell tile per wave)
constexpr int kWavesPerBlock  = 8;                // 256-thread blocks
constexpr int kBlocks         = kTiles / kWavesPerBlock; // 1024 (32 blocks/batch)
constexpr int kChunks         = kH / 8;           // 32 eight-element K-chunks

// ---------------------------------------------------------------------------
// Kernel 1: normalize pos_query per batch into ws, zero the accumulators.
// ---------------------------------------------------------------------------
__global__ __launch_bounds__(256) void cl_init(const float* __restrict__ q,
                                               float* __restrict__ qn,
                                               float* f_num, float* f_den,
                                               int* i_cp, int* i_cn) {
    int b = blockIdx.x;          // 32 blocks
    int t = threadIdx.x;         // 256 threads = H
    float v = q[b * kH + t];
    float p = v * v;
    #pragma unroll
    for (int off = 16; off; off >>= 1) p += __shfl_xor(p, off, 32);
    __shared__ float wsum[8];
    if ((t & 31) == 0) wsum[t >> 5] = p;
    __syncthreads();
    __shared__ float snorm;
    if (t == 0) {
        float s = 0.f;
        #pragma unroll
        for (int w = 0; w < 8; ++w) s += wsum[w];
        snorm = sqrtf(s) + CL_EPS;
    }
    __syncthreads();
    qn[b * kH + t] = v / snorm;
    if (b == 0 && t < kB) { f_num[t] = 0.f; f_den[t] = 0.f; i_cp[t] = 0; i_cn[t] = 0; }
}

// ---------------------------------------------------------------------------
// Kernel 2: per-tile WMMA dot products + squared norms + masked exp-sums.
// ---------------------------------------------------------------------------
__global__ __launch_bounds__(256) void cl_main(const float* __restrict__ tmap,
                                               const float* __restrict__ qn,
                                               const unsigned char* __restrict__ mpos,
                                               const unsigned char* __restrict__ mneg,
                                               float* f_num, float* f_den,
                                               int* i_cp, int* i_cn) {
    const int lane = threadIdx.x & 31;
    const int wave = threadIdx.x >> 5;
    const int tile = blockIdx.x * kWavesPerBlock + wave;    // < kTiles
    const int b    = (tile * 16) / kCellsPerBatch;          // batch (uniform per block)
    const int half = lane >> 4;                             // K-slot group select
    const int row  = lane & 15;                             // cell within tile

    // Per 8-element chunk c: lane<16 loads elems {8c..8c+3}, lane>=16 loads
    // {8c+4..8c+7}. WMMA #1 consumes dwords {0,1} of the quad, WMMA #2 dwords
    // {2,3}; A and B use the identical mapping, so every element is counted
    // exactly once against its q partner (reduction order is irrelevant).
    const float* __restrict__ arow = tmap + (size_t)(tile * 16 + row) * kH + half * 4;
    const float* __restrict__ brow = qn + b * kH + half * 4;

    v8f acc = {};          // 16x16 f32 accumulator: D[m][n] = dot(cell m, q) for all n
    float sq = 0.f;        // per-lane partial of sum(t^2) for cell `row`, this K-subset

    // Software pipeline, prefetch distance 2: ~6 b128 loads in flight per wave.
    v4f a0 = *(const v4f*)(arow + 0 * 8);
    v4f b0 = *(const v4f*)(brow + 0 * 8);
    v4f a1 = *(const v4f*)(arow + 1 * 8);
    v4f b1 = *(const v4f*)(brow + 1 * 8);

    #pragma unroll 4
    for (int c = 0; c < kChunks - 2; ++c) {
        v4f a2 = *(const v4f*)(arow + (c + 2) * 8);
        v4f b2 = *(const v4f*)(brow + (c + 2) * 8);
        sq = fmaf(a0.x, a0.x, fmaf(a0.y, a0.y, fmaf(a0.z, a0.z, fmaf(a0.w, a0.w, sq))));
        acc = __builtin_amdgcn_wmma_f32_16x16x4_f32(
                  false, __builtin_shufflevector(a0, a0, 0, 1),
                  false, __builtin_shufflevector(b0, b0, 0, 1),
                  (short)0, acc, false, false);
        acc = __builtin_amdgcn_wmma_f32_16x16x4_f32(
                  false, __builtin_shufflevector(a0, a0, 2, 3),
                  false, __builtin_shufflevector(b0, b0, 2, 3),
                  (short)0, acc, false, false);
        a0 = a1; b0 = b1;
        a1 = a2; b1 = b2;
    }
    // Drain the pipeline.
    #pragma unroll
    for (int t = 0; t < 2; ++t) {
        sq = fmaf(a0.x, a0.x, fmaf(a0.y, a0.y, fmaf(a0.z, a0.z, fmaf(a0.w, a0.w, sq))));
        acc = __builtin_amdgcn_wmma_f32_16x16x4_f32(
                  false, __builtin_shufflevector(a0, a0, 0, 1),
                  false, __builtin_shufflevector(b0, b0, 0, 1),
                  (short)0, acc, false, false);
        acc = __builtin_amdgcn_wmma_f32_16x16x4_f32(
                  false, __builtin_shufflevector(a0, a0, 2, 3),
                  false, __builtin_shufflevector(b0, b0, 2, 3),
                  (short)0, acc, false, false);
        a0 = a1; b0 = b1;
    }

    // Combine the two K-subsets: lanes m and m+16 both end with full ||cell m||^2.
    sq += __shfl_xor(sq, 16, 32);

    // D layout: VGPR k holds cell k on lanes 0-15 and cell k+8 on lanes 16-31.
    // Lane 0 finalizes cells 0..7, lane 16 finalizes cells 8..15.
    float lnum = 0.f, lden = 0.f;
    int   lcp = 0, lcn = 0;
    const int cellBase = tile * 16 + (half << 3);
    const int sqSrcAdd = (lane >> 1) & 8;   // lane0 -> 0, lane16 -> 8
    #pragma unroll
    for (int k = 0; k < 8; ++k) {
        float sqk = __shfl(sq, k + sqSrcAdd, 32);   // all lanes participate
        if ((lane & 15) == 0) {
            float d    = acc[k];
            int   cell = cellBase + k;
            float nt   = sqrtf(sqk);
            // u1 = t/||t||; ||u1|| == 1 -> s = (t.q)/(||t||*(1+eps))
            float s = d / (nt * (1.0f + CL_EPS));
            float e = expf(s);                      // TAO = 1
            if (mpos[cell]) { lnum += e; lcp++; }
            if (mneg[cell]) { lden += e; lcn++; }
        }
    }

    // Wave reduce (only lanes 0 and 16 hold nonzero partials).
    #pragma unroll
    for (int off = 16; off; off >>= 1) {
        lnum += __shfl_xor(lnum, off, 32);
        lden += __shfl_xor(lden, off, 32);
        lcp  += __shfl_xor(lcp,  off, 32);
        lcn  += __shfl_xor(lcn,  off, 32);
    }

    __shared__ float s_n[kWavesPerBlock], s_d[kWavesPerBlock];
    __shared__ int   s_p[kWavesPerBlock], s_q[kWavesPerBlock];
    if (lane == 0) { s_n[wave] = lnum; s_d[wave] = lden; s_p[wave] = lcp; s_q[wave] = lcn; }
    __syncthreads();
    if (threadIdx.x == 0) {
        float tn = 0.f, td = 0.f; int tp = 0, tq = 0;
        #pragma unroll
        for (int w = 0; w < kWavesPerBlock; ++w) { tn += s_n[w]; td += s_d[w]; tp += s_p[w]; tq += s_q[w]; }
        // Block covers 8 consecutive tiles -> single batch -> 4 atomics/block.
        atomicAdd(&f_num[b], tn);
        atomicAdd(&f_den[b], td);
        atomicAdd(&i_cp[b], tp);
        atomicAdd(&i_cn[b], tq);
    }
}

// ---------------------------------------------------------------------------
// Kernel 3: per-batch InfoNCE loss, validity mask, mean over valid samples.
// ---------------------------------------------------------------------------
__global__ void cl_final(const float* f_num, const float* f_den,
                         const int* i_cp, const int* i_cn, float* out) {
    int lane = threadIdx.x;    // 32 threads
    float li = 0.f; int v = 0;
    if (lane < kB) {
        float num = f_num[lane];
        float neg = f_den[lane];
        if (i_cp[lane] > 0 && i_cn[lane] > 0) {
            float den = num + neg;
            li = -logf(num / (den + CL_EPS));
            v = 1;
        }
    }
    #pragma unroll
    for (int off = 16; off; off >>= 1) {
        li += __shfl_xor(li, off, 32);
        v  += __shfl_xor(v,  off, 32);
    }
    if (lane == 0) out[0] = li / (float)(v > 0 ? v : 1);
}

// ---------------------------------------------------------------------------
extern "C" void kernel_launch(void* const* d_in, const int* in_sizes, int n_in,
                              void* d_out, int out_size, void* d_ws, size_t ws_size,
                              hipStream_t stream) {
    (void)in_sizes; (void)n_in; (void)out_size; (void)ws_size;
    const float*         pos_query = (const float*)d_in[0];         // (B,H) f32
    const float*         tmap      = (const float*)d_in[1];         // (B,S,S,H) f32
    const unsigned char* mpos      = (const unsigned char*)d_in[2]; // (B,S,S) bool
    const unsigned char* mneg      = (const unsigned char*)d_in[3]; // (B,S,S) bool
    float* out = (float*)d_out;

    // Workspace layout: num[32] f32 | den[32] f32 | cp[32] i32 | cn[32] i32 | qn[32*256] f32
    float* f_num = (float*)d_ws;
    float* f_den = f_num + kB;
    int*   i_cp  = (int*)(f_den + kB);
    int*   i_cn  = i_cp + kB;
    float* qn    = (float*)(i_cn + kB);   // 32 KiB

    cl_init <<<kB, kH, 0, stream>>>(pos_query, qn, f_num, f_den, i_cp, i_cn);
    cl_main <<<kBlocks, 32 * kWavesPerBlock, 0, stream>>>(tmap, qn, mpos, mneg,
                                                          f_num, f_den, i_cp, i_cn);
    cl_final<<<1, 32, 0, stream>>>(f_num, f_den, i_cp, i_cn, out);
}